// LSTM_68075231641639
// MI455X (gfx1250) — compile-verified
//
#include <hip/hip_runtime.h>
#include <hip/hip_bf16.h>

// Problem constants (match reference)
#define B_    4096
#define T_    200
#define D_    4
#define H_    2
#define N_    (B_ * T_)        // 819200 rows into the MLP head
#define EPS_  1e-5f
#define SLOPE_ 0.01f

typedef __attribute__((ext_vector_type(2))) float v2f;
typedef __attribute__((ext_vector_type(8))) float v8f;

__device__ __forceinline__ float sigf(float x) { return 1.0f / (1.0f + expf(-x)); }
__device__ __forceinline__ float lrelu(float x) { return x >= 0.0f ? x : SLOPE_ * x; }

// ---------------------------------------------------------------------------
// Zero the stats scratch region
// ---------------------------------------------------------------------------
__global__ void k_zero(float* __restrict__ p, int n) {
    int i = blockIdx.x * blockDim.x + threadIdx.x;
    if (i < n) p[i] = 0.0f;
}

// ---------------------------------------------------------------------------
// LSTM: one thread per batch row, 200 sequential steps.
// Writes h[b,t,0:2] to hbuf (row n = b*T + t), accumulates 5 moments of h
// (for the analytic layer-1 BN statistics).
// ---------------------------------------------------------------------------
__global__ __launch_bounds__(256) void k_lstm(
    const float* __restrict__ x,
    const float* __restrict__ Wih, const float* __restrict__ Whh,
    const float* __restrict__ bih, const float* __restrict__ bhh,
    float* __restrict__ hbuf, float* __restrict__ mom)
{
    int b = blockIdx.x * blockDim.x + threadIdx.x;   // 0..4095

    float wih[8][4], whh[8][2], bs[8];
#pragma unroll
    for (int g = 0; g < 8; ++g) {
#pragma unroll
        for (int d = 0; d < 4; ++d) wih[g][d] = Wih[g * 4 + d];
        whh[g][0] = Whh[g * 2 + 0];
        whh[g][1] = Whh[g * 2 + 1];
        bs[g] = bih[g] + bhh[g];
    }

    float h0 = 0.f, h1 = 0.f, c0 = 0.f, c1 = 0.f;
    float s1 = 0.f, s2 = 0.f, s11 = 0.f, s22 = 0.f, s12 = 0.f;

    const float4* xp = (const float4*)x + (size_t)b * T_;   // D=4 -> one float4/step
    float2* hp = (float2*)hbuf + (size_t)b * T_;

    for (int t = 0; t < T_; ++t) {
        float4 xv = xp[t];
        float z[8];
#pragma unroll
        for (int g = 0; g < 8; ++g) {
            z[g] = bs[g]
                 + wih[g][0] * xv.x + wih[g][1] * xv.y
                 + wih[g][2] * xv.z + wih[g][3] * xv.w
                 + whh[g][0] * h0 + whh[g][1] * h1;
        }
        // PyTorch gate order i,f,g,o; each of width H=2
        float i0 = sigf(z[0]), i1 = sigf(z[1]);
        float f0 = sigf(z[2]), f1 = sigf(z[3]);
        float g0 = tanhf(z[4]), g1 = tanhf(z[5]);
        float o0 = sigf(z[6]), o1 = sigf(z[7]);
        c0 = f0 * c0 + i0 * g0;
        c1 = f1 * c1 + i1 * g1;
        h0 = o0 * tanhf(c0);
        h1 = o1 * tanhf(c1);
        hp[t] = make_float2(h0, h1);
        s1 += h0; s2 += h1; s11 += h0 * h0; s22 += h1 * h1; s12 += h0 * h1;
    }

    __shared__ float red[256];
    float vals[5] = { s1, s2, s11, s22, s12 };
    for (int q = 0; q < 5; ++q) {
        red[threadIdx.x] = vals[q];
        __syncthreads();
        for (int off = 128; off > 0; off >>= 1) {
            if ((int)threadIdx.x < off) red[threadIdx.x] += red[threadIdx.x + off];
            __syncthreads();
        }
        if (threadIdx.x == 0) atomicAdd(&mom[q], red[0]);
        __syncthreads();
    }
}

// ---------------------------------------------------------------------------
// Analytic layer-1 BN stats from the 2x2 moment matrix of h:
//   y1_j = w_j . h + b_j  =>  mean/var derivable from E[h], E[h h^T]
// ---------------------------------------------------------------------------
__global__ void k_fin1(const float* __restrict__ mom,
                       const float* __restrict__ w1, const float* __restrict__ b1f,
                       float* __restrict__ mean1, float* __restrict__ istd1)
{
    int j = threadIdx.x;
    if (j >= 32) return;
    float invN = 1.0f / (float)N_;
    float m1 = mom[0] * invN, m2 = mom[1] * invN;
    float E11 = mom[2] * invN, E22 = mom[3] * invN, E12 = mom[4] * invN;
    float w0 = w1[j * 2 + 0], wq = w1[j * 2 + 1], bb = b1f[j];
    float mu  = w0 * m1 + wq * m2 + bb;
    float Ey2 = w0 * w0 * E11 + 2.f * w0 * wq * E12 + wq * wq * E22
              + 2.f * bb * (w0 * m1 + wq * m2) + bb * bb;
    float var = Ey2 - mu * mu;             // biased (ddof=0), matches jnp.var
    mean1[j] = mu;
    istd1[j] = rsqrtf(var + EPS_);
}

// Generic per-feature finalize: mean = S/N, istd = rsqrt(Sq/N - mean^2 + eps)
__global__ void k_fin(const float* __restrict__ gsum, const float* __restrict__ gsq,
                      float* __restrict__ mean, float* __restrict__ istd, int nf)
{
    int j = threadIdx.x;
    if (j >= nf) return;
    float invN = 1.0f / (float)N_;
    float mu = gsum[j] * invN;
    float var = gsq[j] * invN - mu * mu;
    mean[j] = mu;
    istd[j] = rsqrtf(var + EPS_);
}

// ---------------------------------------------------------------------------
// Per-row layer-1: a1 = lrelu(bn1(h @ W1^T + b1)), one row/thread into LDS.
// ---------------------------------------------------------------------------
__device__ __forceinline__ void compute_a1(
    const float* __restrict__ hbuf, int rowBase,
    const float* __restrict__ w1, const float* __restrict__ b1f,
    const float* __restrict__ mean1, const float* __restrict__ istd1,
    const float* __restrict__ g1, const float* __restrict__ bb1,
    float (*a1)[32])
{
    int tid = threadIdx.x;                                   // 0..127
    float2 hv = ((const float2*)hbuf)[rowBase + tid];
#pragma unroll
    for (int j = 0; j < 32; ++j) {
        float y = w1[j * 2 + 0] * hv.x + w1[j * 2 + 1] * hv.y + b1f[j];
        float a = (y - mean1[j]) * istd1[j] * g1[j] + bb1[j];
        a1[tid][j] = lrelu(a);
    }
}

// ---------------------------------------------------------------------------
// fc2 16x16 output tile via chained V_WMMA_F32_16X16X4_F32 (K = 32 = 8 x 4).
// 32-bit A (16x4): lanes 0-15 hold K={0,1}, lanes 16-31 hold K={2,3} per step.
// 32-bit B (4x16): VGPR0 = K0/K2 rows, VGPR1 = K1/K3; B[k][n] = fc2_w[n][k].
// D: VGPR v -> row v (lanes 0-15) / row v+8 (lanes 16-31), col = lane&15.
// ---------------------------------------------------------------------------
__device__ __forceinline__ v8f wmma_fc2_tile(
    const float (*a1)[32], int rowLocalBase,
    const float* __restrict__ w2, int ct, int lane)
{
    int lid = lane & 15;
    bool hi = lane >= 16;
    int n = ct * 16 + lid;
    v8f c = {};
#pragma unroll
    for (int kk = 0; kk < 8; ++kk) {
        int kb = kk * 4 + (hi ? 2 : 0);
        int arow = rowLocalBase + lid;
        v2f a = { a1[arow][kb], a1[arow][kb + 1] };
        v2f b = { w2[n * 32 + kb], w2[n * 32 + kb + 1] };
        c = __builtin_amdgcn_wmma_f32_16x16x4_f32(
                /*neg_a=*/false, a, /*neg_b=*/false, b,
                /*c_mod=*/(short)0, c, /*reuse_a=*/false, /*reuse_b=*/false);
    }
    return c;
}

// ---------------------------------------------------------------------------
// Pass A: y2 = a1 @ W2^T + b2 via WMMA; accumulate per-feature sum / sumsq.
// 128 rows/block, 4 waves, each wave does 2 row-tiles x 5 col-tiles.
// ---------------------------------------------------------------------------
__global__ __launch_bounds__(128) void k_pass_a(
    const float* __restrict__ hbuf,
    const float* __restrict__ w1, const float* __restrict__ b1f,
    const float* __restrict__ mean1, const float* __restrict__ istd1,
    const float* __restrict__ g1, const float* __restrict__ bb1,
    const float* __restrict__ w2, const float* __restrict__ b2f,
    float* __restrict__ gsum2, float* __restrict__ gsq2)
{
    __shared__ float a1[128][32];
    __shared__ float ssum[80], ssq[80];
    int tid = threadIdx.x;
    if (tid < 80) { ssum[tid] = 0.f; ssq[tid] = 0.f; }

    int rowBase = blockIdx.x * 128;
    compute_a1(hbuf, rowBase, w1, b1f, mean1, istd1, g1, bb1, a1);
    __syncthreads();

    int wave = tid >> 5, lane = tid & 31, lid = lane & 15;
#pragma unroll
    for (int rt = 0; rt < 2; ++rt) {
        int rlb = (wave * 2 + rt) * 16;
#pragma unroll
        for (int ct = 0; ct < 5; ++ct) {
            v8f c = wmma_fc2_tile(a1, rlb, w2, ct, lane);
            int col = ct * 16 + lid;
            float bias = b2f[col];
            float s = 0.f, sq = 0.f;
#pragma unroll
            for (int v = 0; v < 8; ++v) {
                float y = c[v] + bias;
                s += y; sq += y * y;
            }
            atomicAdd(&ssum[col], s);
            atomicAdd(&ssq[col], sq);
        }
    }
    __syncthreads();
    if (tid < 80) {
        atomicAdd(&gsum2[tid], ssum[tid]);
        atomicAdd(&gsq2[tid], ssq[tid]);
    }
}

// ---------------------------------------------------------------------------
// Pass B: recompute y2 (WMMA), normalize -> a2 (LDS), fc3 per row,
// accumulate layer-3 BN stats.
// ---------------------------------------------------------------------------
__global__ __launch_bounds__(128) void k_pass_b(
    const float* __restrict__ hbuf,
    const float* __restrict__ w1, const float* __restrict__ b1f,
    const float* __restrict__ mean1, const float* __restrict__ istd1,
    const float* __restrict__ g1, const float* __restrict__ bb1,
    const float* __restrict__ w2, const float* __restrict__ b2f,
    const float* __restrict__ mean2, const float* __restrict__ istd2,
    const float* __restrict__ g2, const float* __restrict__ bb2,
    const float* __restrict__ w3, const float* __restrict__ b3f,
    float* __restrict__ gsum3, float* __restrict__ gsq3)
{
    __shared__ float a1[128][32];
    __shared__ float a2[128][80];
    __shared__ float ssum[16], ssq[16];
    int tid = threadIdx.x;
    if (tid < 16) { ssum[tid] = 0.f; ssq[tid] = 0.f; }

    int rowBase = blockIdx.x * 128;
    compute_a1(hbuf, rowBase, w1, b1f, mean1, istd1, g1, bb1, a1);
    __syncthreads();

    int wave = tid >> 5, lane = tid & 31, lid = lane & 15;
    bool hiL = lane >= 16;
#pragma unroll
    for (int rt = 0; rt < 2; ++rt) {
        int rlb = (wave * 2 + rt) * 16;
#pragma unroll
        for (int ct = 0; ct < 5; ++ct) {
            v8f c = wmma_fc2_tile(a1, rlb, w2, ct, lane);
            int col = ct * 16 + lid;
            float bias = b2f[col], mu = mean2[col], is = istd2[col];
            float gg = g2[col], bo = bb2[col];
#pragma unroll
            for (int v = 0; v < 8; ++v) {
                float y = (c[v] + bias - mu) * is * gg + bo;
                a2[rlb + v + (hiL ? 8 : 0)][col] = lrelu(y);
            }
        }
    }
    __syncthreads();

    // fc3: one row per thread
    float y3[16];
#pragma unroll
    for (int j = 0; j < 16; ++j) {
        float acc = b3f[j];
        for (int k = 0; k < 80; ++k) acc += w3[j * 80 + k] * a2[tid][k];
        y3[j] = acc;
    }
#pragma unroll
    for (int j = 0; j < 16; ++j) {
        atomicAdd(&ssum[j], y3[j]);
        atomicAdd(&ssq[j], y3[j] * y3[j]);
    }
    __syncthreads();
    if (tid < 16) {
        atomicAdd(&gsum3[tid], ssum[tid]);
        atomicAdd(&gsq3[tid], ssq[tid]);
    }
}

// ---------------------------------------------------------------------------
// Pass C: recompute through fc3, normalize, fc4 + lrelu, write output.
// ---------------------------------------------------------------------------
__global__ __launch_bounds__(128) void k_pass_c(
    const float* __restrict__ hbuf,
    const float* __restrict__ w1, const float* __restrict__ b1f,
    const float* __restrict__ mean1, const float* __restrict__ istd1,
    const float* __restrict__ g1, const float* __restrict__ bb1,
    const float* __restrict__ w2, const float* __restrict__ b2f,
    const float* __restrict__ mean2, const float* __restrict__ istd2,
    const float* __restrict__ g2, const float* __restrict__ bb2,
    const float* __restrict__ w3, const float* __restrict__ b3f,
    const float* __restrict__ mean3, const float* __restrict__ istd3,
    const float* __restrict__ g3, const float* __restrict__ bb3,
    const float* __restrict__ w4, const float* __restrict__ b4f,
    float* __restrict__ out)
{
    __shared__ float a1[128][32];
    __shared__ float a2[128][80];
    int tid = threadIdx.x;
    int rowBase = blockIdx.x * 128;

    compute_a1(hbuf, rowBase, w1, b1f, mean1, istd1, g1, bb1, a1);
    __syncthreads();

    int wave = tid >> 5, lane = tid & 31, lid = lane & 15;
    bool hiL = lane >= 16;
#pragma unroll
    for (int rt = 0; rt < 2; ++rt) {
        int rlb = (wave * 2 + rt) * 16;
#pragma unroll
        for (int ct = 0; ct < 5; ++ct) {
            v8f c = wmma_fc2_tile(a1, rlb, w2, ct, lane);
            int col = ct * 16 + lid;
            float bias = b2f[col], mu = mean2[col], is = istd2[col];
            float gg = g2[col], bo = bb2[col];
#pragma unroll
            for (int v = 0; v < 8; ++v) {
                float y = (c[v] + bias - mu) * is * gg + bo;
                a2[rlb + v + (hiL ? 8 : 0)][col] = lrelu(y);
            }
        }
    }
    __syncthreads();

    // fc3 + bn3 + lrelu, then fc4 + lrelu; one row per thread
    float a3[16];
#pragma unroll
    for (int j = 0; j < 16; ++j) {
        float acc = b3f[j];
        for (int k = 0; k < 80; ++k) acc += w3[j * 80 + k] * a2[tid][k];
        float y = (acc - mean3[j]) * istd3[j] * g3[j] + bb3[j];
        a3[j] = lrelu(y);
    }
    float o0 = b4f[0], o1 = b4f[1];
#pragma unroll
    for (int k = 0; k < 16; ++k) {
        o0 += w4[k] * a3[k];
        o1 += w4[16 + k] * a3[k];
    }
    int n = rowBase + tid;
    out[2 * n + 0] = lrelu(o0);
    out[2 * n + 1] = lrelu(o1);
}

// ---------------------------------------------------------------------------
// Host-side launch sequence (all on `stream`, graph-capture safe).
// ---------------------------------------------------------------------------
extern "C" void kernel_launch(void* const* d_in, const int* in_sizes, int n_in,
                              void* d_out, int out_size, void* d_ws, size_t ws_size,
                              hipStream_t stream)
{
    (void)in_sizes; (void)n_in; (void)out_size; (void)ws_size;

    const float* x   = (const float*)d_in[0];
    const float* Wih = (const float*)d_in[1];
    const float* Whh = (const float*)d_in[2];
    const float* bih = (const float*)d_in[3];
    const float* bhh = (const float*)d_in[4];
    const float* w1  = (const float*)d_in[5];
    const float* b1  = (const float*)d_in[6];
    const float* w2  = (const float*)d_in[7];
    const float* b2  = (const float*)d_in[8];
    const float* w3  = (const float*)d_in[9];
    const float* b3  = (const float*)d_in[10];
    const float* w4  = (const float*)d_in[11];
    const float* b4  = (const float*)d_in[12];
    const float* g1  = (const float*)d_in[13];
    const float* bb1 = (const float*)d_in[14];
    const float* g2  = (const float*)d_in[15];
    const float* bb2 = (const float*)d_in[16];
    const float* g3  = (const float*)d_in[17];
    const float* bb3 = (const float*)d_in[18];
    float* out = (float*)d_out;

    // Workspace layout (floats): h buffer [N,2], then stats block.
    float* wsf   = (float*)d_ws;
    float* hbuf  = wsf;                 // 2*N floats
    float* st    = wsf + 2 * (size_t)N_;
    float* mom   = st + 0;              // 5 moments of h
    float* mean1 = st + 8;   float* istd1 = st + 40;    // 32 each
    float* sum2  = st + 72;  float* sq2   = st + 152;   // 80 each
    float* mean2 = st + 232; float* istd2 = st + 312;   // 80 each
    float* sum3  = st + 392; float* sq3   = st + 408;   // 16 each
    float* mean3 = st + 424; float* istd3 = st + 440;   // 16 each

    k_zero<<<2, 256, 0, stream>>>(st, 512);
    k_lstm<<<B_ / 256, 256, 0, stream>>>(x, Wih, Whh, bih, bhh, hbuf, mom);
    k_fin1<<<1, 32, 0, stream>>>(mom, w1, b1, mean1, istd1);
    k_pass_a<<<N_ / 128, 128, 0, stream>>>(hbuf, w1, b1, mean1, istd1, g1, bb1,
                                           w2, b2, sum2, sq2);
    k_fin<<<1, 80, 0, stream>>>(sum2, sq2, mean2, istd2, 80);
    k_pass_b<<<N_ / 128, 128, 0, stream>>>(hbuf, w1, b1, mean1, istd1, g1, bb1,
                                           w2, b2, mean2, istd2, g2, bb2,
                                           w3, b3, sum3, sq3);
    k_fin<<<1, 16, 0, stream>>>(sum3, sq3, mean3, istd3, 16);
    k_pass_c<<<N_ / 128, 128, 0, stream>>>(hbuf, w1, b1, mean1, istd1, g1, bb1,
                                           w2, b2, mean2, istd2, g2, bb2,
                                           w3, b3, mean3, istd3, g3, bb3,
                                           w4, b4, out);
}